// Screen_53343493816648
// MI455X (gfx1250) — compile-verified
//
#include <hip/hip_runtime.h>
#include <stdint.h>

// Screen constants (match reference)
#define SCR_NX 1024
#define SCR_NY 1024
#define SCR_LEFT  (-5.12e-3f)
#define SCR_BOT   (-5.12e-3f)
#define SCR_RIGHT ( 5.12e-3f)
#define SCR_TOP   ( 5.12e-3f)
#define SCR_INV   (1.0e5f)   // 1/HSTEP = 1/VSTEP

__global__ void screen_zero(float* __restrict__ out, int n) {
    int i = blockIdx.x * blockDim.x + threadIdx.x;
    if (i < n) out[i] = 0.0f;
}

__device__ __forceinline__ void deposit(float px, float py, float m0, float m1,
                                        float* __restrict__ out) {
    float x = px - m0;
    float y = py - m1;
    int ix = (int)floorf((x - SCR_LEFT) * SCR_INV);
    int iy = (int)floorf((y - SCR_BOT)  * SCR_INV);
    ix = (x == SCR_RIGHT) ? (SCR_NX - 1) : ix;
    iy = (y == SCR_TOP)   ? (SCR_NY - 1) : iy;
    bool valid = (x >= SCR_LEFT) && (x <= SCR_RIGHT) &&
                 (y >= SCR_BOT)  && (y <= SCR_TOP);
    ix = min(max(ix, 0), SCR_NX - 1);
    iy = min(max(iy, 0), SCR_NY - 1);
    if (valid) {
        // image = flipud(hist.T): image[NY-1-iy, ix]
        atomicAdd(&out[(SCR_NY - 1 - iy) * SCR_NX + ix], 1.0f);
    }
}

// Issue one tile's async copies: this lane's 16B of xs and 16B of ys into its
// wave-private LDS slots (GVS mode: saddr = array base, vaddr = byte offset).
// NT hint: the 128MB particle stream is read once; don't evict the
// L2-resident 4MB image.
__device__ __forceinline__ void issue_tile(unsigned ldsx, unsigned ldsy,
                                           unsigned voff,
                                           const float* xs, const float* ys) {
    asm volatile(
        "global_load_async_to_lds_b128 %0, %2, %3 th:TH_LOAD_NT\n\t"
        "global_load_async_to_lds_b128 %1, %2, %4 th:TH_LOAD_NT"
        :
        : "v"(ldsx), "v"(ldsy), "v"(voff), "s"(xs), "s"(ys)
        : "memory");
}

// Double-buffered wave-private async staging. Each wave32 stages 128 xs +
// 128 ys per tile; while it deposits tile i it already has tile i+1's async
// copies in flight (async loads complete in order, so s_wait_asynccnt 2 ==
// "oldest pair done, newest pair may still fly").
//
// NOTE: all LDS readbacks index the __shared__ array directly (no generic
// float* to it is ever formed), so they stay addrspace(3) -> ds_load_b128.
__global__ __launch_bounds__(256) void screen_hist(
    const float* __restrict__ xs, const float* __restrict__ ys,
    const float* __restrict__ mis, float* __restrict__ out, int n) {
    __shared__ float stage[4096];  // 2 buffers * 8 waves * 256 floats = 16 KB

    const int lane = threadIdx.x & 31;
    const int wave = threadIdx.x >> 5;
    const int xslot = wave * 256 + lane * 4;  // lane's x float4 slot (buffer 0)
    const int yslot = xslot + 128;            // lane's y float4 slot (buffer 0)

    // Raw LDS byte addresses for the async-copy destinations (plain scalars;
    // generic shared-pointer low 32 bits == LDS byte offset since the shared
    // aperture lives in addr[63:32]).
    const unsigned lx0 = (unsigned)(uintptr_t)(&stage[xslot]);
    const unsigned ly0 = (unsigned)(uintptr_t)(&stage[yslot]);
    const unsigned lx1 = (unsigned)(uintptr_t)(&stage[xslot + 2048]);
    const unsigned ly1 = (unsigned)(uintptr_t)(&stage[yslot + 2048]);

    const float m0 = mis[0];
    const float m1 = mis[1];

    const long long wavesTotal = (long long)gridDim.x * (blockDim.x >> 5);
    const long long stride     = wavesTotal * 128;
    const long long gwave      = (long long)blockIdx.x * (blockDim.x >> 5) + wave;
    const long long laneByte   = (long long)lane * 16;  // lane's byte offset in tile

    long long base = gwave * 128;

    if (base + 128 <= (long long)n) {
        // Prime the pipeline with tile 0 into buffer 0.
        issue_tile(lx0, ly0, (unsigned)(base * 4 + laneByte), xs, ys);
        int cur = 0;
        for (;;) {
            long long next = base + stride;
            bool hasNext = (next + 128 <= (long long)n);
            if (hasNext) {
                // Prefetch next tile into the other buffer, then wait only
                // for the *current* tile's pair (oldest two async ops).
                issue_tile(cur ? lx0 : lx1, cur ? ly0 : ly1,
                           (unsigned)(next * 4 + laneByte), xs, ys);
                asm volatile("s_wait_asynccnt 2" ::: "memory");
            } else {
                asm volatile("s_wait_asynccnt 0" ::: "memory");
            }

            // Direct shared-array indexing -> ds_load_b128 (addrspace(3)).
            const int bufIdx = cur ? 2048 : 0;
            float4 xv = *(const float4*)(&stage[bufIdx + xslot]);
            float4 yv = *(const float4*)(&stage[bufIdx + yslot]);
            deposit(xv.x, yv.x, m0, m1, out);
            deposit(xv.y, yv.y, m0, m1, out);
            deposit(xv.z, yv.z, m0, m1, out);
            deposit(xv.w, yv.w, m0, m1, out);

            base = next;
            if (!hasNext) break;
            cur ^= 1;
        }
    }

    // Guarded scalar tail (never hit for N = 16.7M with this launch config,
    // kept for generality).
    for (; base < (long long)n; base += stride) {
        for (int k = 0; k < 4; ++k) {
            long long idx = base + (long long)lane * 4 + k;
            if (idx < (long long)n) deposit(xs[idx], ys[idx], m0, m1, out);
        }
    }
}

extern "C" void kernel_launch(void* const* d_in, const int* in_sizes, int n_in,
                              void* d_out, int out_size, void* d_ws, size_t ws_size,
                              hipStream_t stream) {
    const float* xs  = (const float*)d_in[0];
    const float* ys  = (const float*)d_in[1];
    const float* mis = (const float*)d_in[2];
    float* out = (float*)d_out;
    int n    = in_sizes[0];
    int npix = out_size;  // 1024*1024

    // d_out is poisoned by the harness; zero it every call (deterministic).
    screen_zero<<<(npix + 255) / 256, 256, 0, stream>>>(out, npix);

    // 2048 blocks * 8 wave32 = 16384 waves; 128 particles/wave/tile ->
    // 2.097M particles per sweep, 8 pipelined sweeps over 16.7M.
    screen_hist<<<2048, 256, 0, stream>>>(xs, ys, mis, out, n);
}